// RPN_60662118089263
// MI455X (gfx1250) — compile-verified
//
#include <hip/hip_runtime.h>
#include <hip/hip_bf16.h>
#include <stdint.h>

// ---------------------------------------------------------------------------
// RPN proposal filtering for MI455X (gfx1250, wave32, 320KB LDS per WGP).
// Latency/selection-bound problem: radix-select topk, on-chip bitonic sort,
// tiled IoU bitmask NMS with (optional) async global->LDS tile loads.
// ---------------------------------------------------------------------------

#define NLEV 5
#define SEL 4507                 // 1000+1000+1000+1000+507
#define MW 141                   // ceil(SEL/32)
#define NSORT 8192               // pow2 >= SEL
#define POST_N 1000
#define NMS_TH 0.7f
#define MIN_SZ 0.001f
#define RS_NT 1024

__device__ __constant__ int c_lvl_off[NLEV]  = {0, 120000, 150000, 157500, 159375};
__device__ __constant__ int c_lvl_n[NLEV]    = {120000, 30000, 7500, 1875, 507};
__device__ __constant__ int c_lvl_k[NLEV]    = {1000, 1000, 1000, 1000, 507};
__device__ __constant__ int c_sel_base[NLEV] = {0, 1000, 2000, 3000, 4000};

#define AS1 __attribute__((address_space(1)))
#define AS3 __attribute__((address_space(3)))
typedef int v4i __attribute__((vector_size(16)));

#if defined(__has_builtin)
#if __has_builtin(__builtin_amdgcn_global_load_async_to_lds_b128) && \
    __has_builtin(__builtin_amdgcn_s_wait_asynccnt)
#define USE_ASYNC_LDS 1
#endif
#endif
#ifndef USE_ASYNC_LDS
#define USE_ASYNC_LDS 0
#endif

__device__ __forceinline__ unsigned keyOf(float f) {
    // Monotonic float->uint transform: bigger key <=> bigger float.
    unsigned u = __float_as_uint(f);
    return (u & 0x80000000u) ? ~u : (u | 0x80000000u);
}

// ------------------------------ d_out zeroing ------------------------------
__global__ void zero_kernel(float* __restrict__ out, int n) {
    int i = blockIdx.x * blockDim.x + threadIdx.x;
    if (i < n) out[i] = 0.0f;
}

// --------------------- radix-select top-k per (level, b) -------------------
// One 1024-thread block per (level, image). 4 byte-passes of histogramming
// (LDS hist) pin down the exact k-th key T; then a deterministic, stable
// block-scan compaction emits indices of keys > T, then index-ordered ties.
__global__ __launch_bounds__(RS_NT)
void rs_kernel(const float* __restrict__ obj, int A, int* __restrict__ selIdx) {
    const int lvl = blockIdx.x, b = blockIdx.y;
    const int n = c_lvl_n[lvl], k = c_lvl_k[lvl];
    const int base = c_lvl_off[lvl], selbase = c_sel_base[lvl];
    const float* s = obj + (size_t)b * A + base;
    const int tid = threadIdx.x;

    __shared__ unsigned hist[256];
    __shared__ unsigned sh_byte, sh_gtAdd;
    __shared__ unsigned scanbuf[RS_NT];
    __shared__ unsigned baseG, baseE;

    unsigned prefix = 0, kRem = (unsigned)k;
    for (int p = 0; p < 4; ++p) {
        const int shift = 24 - 8 * p;
        const unsigned maskHi = p ? (0xFFFFFFFFu << (32 - 8 * p)) : 0u;
        if (tid < 256) hist[tid] = 0u;
        __syncthreads();
        for (int i = tid; i < n; i += RS_NT) {
            if (i + 4 * RS_NT < n) __builtin_prefetch(&s[i + 4 * RS_NT], 0, 0);
            unsigned u = keyOf(s[i]);
            if ((u & maskHi) == prefix) atomicAdd(&hist[(u >> shift) & 255u], 1u);
        }
        __syncthreads();
        if (tid == 0) {
            unsigned acc = 0, byte = 0, gtAdd = 0;
            for (int bb = 255; bb >= 0; --bb) {
                unsigned h = hist[bb];
                if (acc + h >= kRem) { byte = (unsigned)bb; gtAdd = acc; break; }
                acc += h;
            }
            sh_byte = byte; sh_gtAdd = gtAdd;
        }
        __syncthreads();
        prefix |= (sh_byte << shift);
        kRem -= sh_gtAdd;
        __syncthreads();
    }
    const unsigned T = prefix;          // exact k-th largest key
    const unsigned E = kRem;            // how many ties at T we need
    const unsigned Gtot = (unsigned)k - E;

    if (tid == 0) { baseG = 0; baseE = 0; }
    __syncthreads();

    int* out = selIdx + (size_t)b * SEL + selbase;
    for (int i0 = 0; i0 < n; i0 += RS_NT) {
        const int i = i0 + tid;
        const bool inR = (i < n);
        const unsigned u = inR ? keyOf(s[i]) : 0u;
        const bool isG = inR && (u > T);
        const bool isE = inR && (u == T);
        const unsigned v = (isG ? 1u : 0u) | (isE ? 0x10000u : 0u);
        scanbuf[tid] = v;
        __syncthreads();
        for (int off = 1; off < RS_NT; off <<= 1) {  // packed inclusive scan
            unsigned mine = scanbuf[tid];
            unsigned add = (tid >= off) ? scanbuf[tid - off] : 0u;
            __syncthreads();
            scanbuf[tid] = mine + add;
            __syncthreads();
        }
        const unsigned incl = scanbuf[tid];
        const unsigned tot = scanbuf[RS_NT - 1];
        const unsigned excl = incl - v;
        if (isG) out[baseG + (excl & 0xFFFFu)] = base + i;
        if (isE) {
            unsigned pos = Gtot + baseE + (excl >> 16);
            if (pos < (unsigned)k) out[pos] = base + i;   // index-ordered ties
        }
        __syncthreads();
        if (tid == 0) { baseG += (tot & 0xFFFFu); baseE += (tot >> 16); }
        __syncthreads();
    }
}

// ------------- decode + clip + validity + on-chip bitonic sort -------------
// One block per image; everything lives in LDS (~138KB: only possible with
// CDNA5's 320KB-per-WGP LDS). Sort (score desc, payload=candidate slot).
__global__ __launch_bounds__(1024)
void sd_kernel(const float* __restrict__ pred, const float* __restrict__ obj,
               const float* __restrict__ anch, const float* __restrict__ vsz,
               const int* __restrict__ selIdx,
               float* __restrict__ sbox, float* __restrict__ sscore,
               int* __restrict__ slvl, float* __restrict__ maxc, int A) {
    extern __shared__ unsigned char smem[];
    float* keys   = (float*)smem;                       // NSORT floats
    unsigned* pay = (unsigned*)(smem + NSORT * 4);      // NSORT uints
    float* ubox   = (float*)(smem + NSORT * 8);         // SEL*4 floats
    __shared__ int sMax;

    const int b = blockIdx.x, tid = threadIdx.x, NT = blockDim.x;
    if (tid == 0) sMax = 0;
    __syncthreads();

    const float w = vsz[b * 2 + 0], h = vsz[b * 2 + 1];
    int lmax = 0;
    for (int i = tid; i < NSORT; i += NT) {
        float key = -__builtin_inff();
        if (i < SEL) {
            const int gi = selIdx[(size_t)b * SEL + i];
            const float ax1 = anch[gi * 4 + 0], ay1 = anch[gi * 4 + 1];
            const float ax2 = anch[gi * 4 + 2], ay2 = anch[gi * 4 + 3];
            const float aw = ax2 - ax1, ah = ay2 - ay1;
            const float acx = ax1 + 0.5f * aw, acy = ay1 + 0.5f * ah;
            const float* p = pred + ((size_t)b * A + gi) * 4;
            const float cx = fmaf(p[0], aw, acx), cy = fmaf(p[1], ah, acy);
            const float bw = expf(p[2]) * aw, bh = expf(p[3]) * ah;
            float x1 = cx - 0.5f * bw, y1 = cy - 0.5f * bh;
            float x2 = x1 + bw, y2 = y1 + bh;
            x1 = fminf(fmaxf(x1, 0.f), w); x2 = fminf(fmaxf(x2, 0.f), w);
            y1 = fminf(fmaxf(y1, 0.f), h); y2 = fminf(fmaxf(y2, 0.f), h);
            ubox[i * 4 + 0] = x1; ubox[i * 4 + 1] = y1;
            ubox[i * 4 + 2] = x2; ubox[i * 4 + 3] = y2;
            const float m = fmaxf(fmaxf(x1, y1), fmaxf(x2, y2));
            const int mi = __float_as_int(m);          // coords >= 0
            lmax = lmax > mi ? lmax : mi;
            const bool valid = (x2 - x1 > MIN_SZ) && (y2 - y1 > MIN_SZ);
            const float sc = obj[(size_t)b * A + gi];
            key = valid ? sc : -__builtin_inff();
        }
        keys[i] = key;
        pay[i] = (unsigned)i;
    }
    atomicMax(&sMax, lmax);

    // Bitonic sort, descending, 8 elements/thread in LDS.
    for (unsigned size = 2; size <= NSORT; size <<= 1) {
        for (unsigned stride = size >> 1; stride > 0; stride >>= 1) {
            __syncthreads();
            for (unsigned t = tid; t < NSORT / 2; t += NT) {
                const unsigned pos = 2 * t - (t & (stride - 1));
                const bool asc = (pos & size) != 0;
                const float ka = keys[pos], kc = keys[pos + stride];
                if ((ka < kc) != asc) {
                    keys[pos] = kc; keys[pos + stride] = ka;
                    const unsigned tp = pay[pos];
                    pay[pos] = pay[pos + stride]; pay[pos + stride] = tp;
                }
            }
        }
    }
    __syncthreads();

    if (tid == 0) maxc[b] = __int_as_float(sMax) + 1.0f;

    for (int j = tid; j < SEL; j += NT) {
        const unsigned p = pay[j];
        const bool real = (p < SEL);
        #pragma unroll
        for (int c = 0; c < 4; ++c)
            sbox[((size_t)b * SEL + j) * 4 + c] = real ? ubox[p * 4 + c] : 0.0f;
        sscore[(size_t)b * SEL + j] = keys[j];
        slvl[(size_t)b * SEL + j] = real ? (int)(p / 1000u) : 0;   // 0..4
    }
}

// --------------------- tiled IoU suppression bitmask -----------------------
// One wave per (32-row, 32-col, image) tile. Column boxes staged in LDS via
// CDNA5 async global->LDS loads (ASYNCcnt) when the builtin is available.
__global__ __launch_bounds__(32)
void iou_kernel(const float* __restrict__ sbox, const int* __restrict__ slvl,
                const float* __restrict__ maxc, unsigned* __restrict__ mask) {
    const int cb = blockIdx.x, rb = blockIdx.y, b = blockIdx.z;
    const int t = threadIdx.x;
    __shared__ float tile[32 * 4];
    __shared__ float toff[32];
    const float mc = maxc[b];
    const int col = cb * 32 + t;
    if (col < SEL) {
        const float* gp = sbox + ((size_t)b * SEL + col) * 4;
#if USE_ASYNC_LDS
        __builtin_amdgcn_global_load_async_to_lds_b128(
            (AS1 v4i*)((void*)gp), (AS3 v4i*)((void*)&tile[t * 4]), 0, 0);
#else
        tile[t * 4 + 0] = gp[0]; tile[t * 4 + 1] = gp[1];
        tile[t * 4 + 2] = gp[2]; tile[t * 4 + 3] = gp[3];
#endif
        toff[t] = (float)slvl[(size_t)b * SEL + col] * mc;  // level offset
    } else {
        toff[t] = 0.0f;
    }
#if USE_ASYNC_LDS
    __builtin_amdgcn_s_wait_asynccnt(0);
#endif
    __syncthreads();

    const int row = rb * 32 + t;
    if (row >= SEL) return;
    const float roff = (float)slvl[(size_t)b * SEL + row] * mc;
    const float* rp = sbox + ((size_t)b * SEL + row) * 4;
    const float rx1 = rp[0] + roff, ry1 = rp[1] + roff;
    const float rx2 = rp[2] + roff, ry2 = rp[3] + roff;
    const float rarea = (rx2 - rx1) * (ry2 - ry1);
    unsigned word = 0u;
    #pragma unroll 4
    for (int j = 0; j < 32; ++j) {
        const int cj = cb * 32 + j;
        if (cj >= SEL || cj == row) continue;
        const float co = toff[j];
        const float cx1 = tile[j * 4 + 0] + co, cy1 = tile[j * 4 + 1] + co;
        const float cx2 = tile[j * 4 + 2] + co, cy2 = tile[j * 4 + 3] + co;
        const float xx1 = fmaxf(rx1, cx1), yy1 = fmaxf(ry1, cy1);
        const float xx2 = fminf(rx2, cx2), yy2 = fminf(ry2, cy2);
        const float inter = fmaxf(xx2 - xx1, 0.f) * fmaxf(yy2 - yy1, 0.f);
        const float carea = (cx2 - cx1) * (cy2 - cy1);
        const float iou = inter / (rarea + carea - inter + 1e-9f);
        if (iou > NMS_TH) word |= (1u << j);
    }
    mask[((size_t)b * SEL + row) * MW + cb] = word;
}

// -------------------- serial greedy scan + compaction ----------------------
// One wave per image. 141-word removal vector lives in LDS; kept boxes are
// written straight into d_out (boxes region then scores region).
__global__ __launch_bounds__(32)
void nms_kernel(const float* __restrict__ sbox, const float* __restrict__ sscore,
                const unsigned* __restrict__ mask, float* __restrict__ out, int bs) {
    const int b = blockIdx.x, t = threadIdx.x;
    __shared__ unsigned remv[MW];
    __shared__ int sCnt;
    for (int w = t; w < MW; w += 32) remv[w] = 0u;
    if (t == 0) sCnt = 0;
    __syncthreads();
    const int scoreBase = bs * POST_N * 4;
    for (int i = 0; i < SEL; ++i) {
        const unsigned bit = (remv[i >> 5] >> (i & 31)) & 1u;
        const float sc = sscore[(size_t)b * SEL + i];
        const bool keep = (!bit) && (sc > -__builtin_inff());
        if (keep) {
            const unsigned* row = mask + ((size_t)b * SEL + i) * MW;
            for (int w = t; w < MW; w += 32) remv[w] |= row[w];
            const int c = sCnt;  // uniform across the wave
            if (c < POST_N) {
                if (t < 4) out[((size_t)(b * POST_N + c)) * 4 + t] =
                               sbox[((size_t)b * SEL + i) * 4 + t];
                if (t == 4) out[scoreBase + b * POST_N + c] = sc;
            }
            if (t == 0) sCnt = c + 1;
        }
        __syncthreads();
        if (sCnt >= POST_N) break;
    }
}

// ---------------------------------------------------------------------------
extern "C" void kernel_launch(void* const* d_in, const int* in_sizes, int n_in,
                              void* d_out, int out_size, void* d_ws, size_t ws_size,
                              hipStream_t stream) {
    const float* predicts   = (const float*)d_in[0];  // (bs, A, 4)
    const float* objectness = (const float*)d_in[1];  // (bs, A, 1)
    const float* anchors    = (const float*)d_in[2];  // (A, 4)
    const float* valid_size = (const float*)d_in[3];  // (bs, 2)
    const int bs = in_sizes[3] / 2;
    const int A  = in_sizes[2] / 4;

    uint8_t* ws = (uint8_t*)d_ws;
    size_t off = 0;
    auto carve = [&](size_t bytes) -> void* {
        void* p = ws + off;
        off = (off + bytes + 255) & ~(size_t)255;
        return p;
    };
    int*      selIdx = (int*)     carve((size_t)bs * SEL * 4);
    float*    sbox   = (float*)   carve((size_t)bs * SEL * 4 * 4);
    float*    sscore = (float*)   carve((size_t)bs * SEL * 4);
    int*      slvl   = (int*)     carve((size_t)bs * SEL * 4);
    float*    maxc   = (float*)   carve((size_t)bs * 4);
    unsigned* mask   = (unsigned*)carve((size_t)bs * SEL * MW * 4);
    (void)ws_size; (void)n_in;

    zero_kernel<<<(out_size + 255) / 256, 256, 0, stream>>>((float*)d_out, out_size);

    rs_kernel<<<dim3(NLEV, bs), RS_NT, 0, stream>>>(objectness, A, selIdx);

    const size_t smem = (size_t)NSORT * 8 + (size_t)SEL * 16 + 64;  // ~138KB LDS
    sd_kernel<<<bs, 1024, smem, stream>>>(predicts, objectness, anchors, valid_size,
                                          selIdx, sbox, sscore, slvl, maxc, A);

    iou_kernel<<<dim3(MW, MW, bs), 32, 0, stream>>>(sbox, slvl, maxc, mask);

    nms_kernel<<<bs, 32, 0, stream>>>(sbox, sscore, mask, (float*)d_out, bs);
}